// Pooling_8151847928044
// MI455X (gfx1250) — compile-verified
//
#include <hip/hip_runtime.h>

typedef __attribute__((ext_vector_type(2))) float v2f;
typedef __attribute__((ext_vector_type(8))) float v8f;

#define NPED     8192
#define HID      128
#define NN       36          // 6x6 occupancy bins
#define GRIDN    6
#define T1       256         // phase-1 block size
#define J_CHUNK  1024        // j-range per y-block
#define J_BLOCKS (NPED / J_CHUNK)
#define CNT_STRIDE 37        // 37 coprime with 64 banks -> conflict-free lane spread
#define SELF_BIN (3 * GRIDN + 3)   // rel=(3,3) -> bin 21

// ---------------- phase 0: zero the grid workspace ----------------
__global__ void zero_kernel(float* __restrict__ p, int n) {
    int i = blockIdx.x * blockDim.x + threadIdx.x;
    if (i < n) p[i] = 0.0f;
}

// ---------------- phase 1: pairwise occupancy histogram ----------------
// One thread per pedestrian i; j swept in LDS tiles of 256; per-thread 36-bin
// counters live in LDS (stride 37 words -> conflict-free). Increments use
// ds_add_f32 (atomicAdd on LDS, no-return, uncontended) instead of a
// load/add/store chain. Self-pair is not tested per iteration: it always hits
// bin 21 when pi is non-NaN, so it is subtracted once at the end (only by the
// j-chunk that contains i). Each block.y handles a 1024-wide j chunk; partials
// merged into the global grid with float atomics.
__global__ __launch_bounds__(T1)
void occupancy_kernel(const float* __restrict__ obs2, float* __restrict__ grid) {
    __shared__ float2 tile[T1];
    __shared__ float  cnt[T1 * CNT_STRIDE];

    const int tid = threadIdx.x;
    const int i   = blockIdx.x * T1 + tid;
    const float2 pi = reinterpret_cast<const float2*>(obs2)[i];

    float* mycnt = &cnt[tid * CNT_STRIDE];
#pragma unroll
    for (int b = 0; b < NN; ++b) mycnt[b] = 0.0f;

    const int j0 = blockIdx.y * J_CHUNK;
    for (int jt = j0; jt < j0 + J_CHUNK; jt += T1) {
        __syncthreads();
        tile[tid] = reinterpret_cast<const float2*>(obs2)[jt + tid];
        __syncthreads();
#pragma unroll 4
        for (int jj = 0; jj < T1; ++jj) {
            const float2 pj = tile[jj];
            // rel in grid coords; NaN inputs fail all compares (matches ref mask)
            const float rx = pj.x - pi.x + (GRIDN * 0.5f);
            const float ry = pj.y - pi.y + (GRIDN * 0.5f);
            const bool ok = (rx >= 0.0f) && (rx < (float)GRIDN) &&
                            (ry >= 0.0f) && (ry < (float)GRIDN);
            if (ok) {
                const int bin = (int)rx * GRIDN + (int)ry;  // floor: rx,ry >= 0
                atomicAdd(&mycnt[bin], 1.0f);               // ds_add_f32, no return
            }
        }
    }

    // Remove the self contribution (counted iff pi.x and pi.y are non-NaN),
    // only in the j-chunk that actually contains i.
    if ((i >= j0) && (i < j0 + J_CHUNK) && (pi.x == pi.x) && (pi.y == pi.y)) {
        atomicAdd(&mycnt[SELF_BIN], -1.0f);
    }

#pragma unroll
    for (int b = 0; b < NN; ++b) {
        const float v = mycnt[b];
        if (v != 0.0f) atomicAdd(&grid[i * NN + b], v);
    }
}

// ---------------- phase 2: out = grid @ W^T + b via V_WMMA_F32_16X16X4_F32 ----
// One wave computes one 16x16 output tile. K=36 -> 9 WMMA(K=4) steps.
// A 16x4 f32 frag: lanes 0-15 hold K={0,1}, lanes 16-31 K={2,3} (2 VGPRs).
// B 4x16 frag symmetric. C/D: VGPR r holds M = r + 8*(lane>=16), N = lane%16.
__global__ __launch_bounds__(256)
void gemm_bias_wmma(const float* __restrict__ grid, const float* __restrict__ W,
                    const float* __restrict__ bias, float* __restrict__ out) {
    const int wave = (int)((blockIdx.x * blockDim.x + threadIdx.x) >> 5);
    const int lane = threadIdx.x & 31;
    const int mt = wave >> 3;           // 512 M-tiles
    const int nt = wave & 7;            // 8 N-tiles
    const int m0 = mt * 16;
    const int n0 = nt * 16;
    const int half = lane >> 4;         // 0: K+{0,1}, 1: K+{2,3}
    const int l    = lane & 15;

    // A row for this lane: grid[m0+l][:], offset by 2*half within each K=4 group
    const float* arow = grid + (size_t)(m0 + l) * NN + 2 * half;
    // B col for this lane: W^T[.][n0+l] = W[n0+l][.] (row-major [128,36])
    const float* brow = W + (size_t)(n0 + l) * NN + 2 * half;

    v8f acc = {};
#pragma unroll
    for (int k = 0; k < NN; k += 4) {
        const v2f a  = *reinterpret_cast<const v2f*>(arow + k);  // 8B aligned (even offs)
        const v2f bf = *reinterpret_cast<const v2f*>(brow + k);
        acc = __builtin_amdgcn_wmma_f32_16x16x4_f32(
            /*neg_a=*/false, a, /*neg_b=*/false, bf,
            /*c_mod=*/(short)0, acc, /*reuse_a=*/false, /*reuse_b=*/false);
    }

    const float bv = bias[n0 + l];
    const int row0 = m0 + half * 8;
#pragma unroll
    for (int r = 0; r < 8; ++r) {
        out[(size_t)(row0 + r) * HID + (n0 + l)] = acc[r] + bv;
    }
}

extern "C" void kernel_launch(void* const* d_in, const int* in_sizes, int n_in,
                              void* d_out, int out_size, void* d_ws, size_t ws_size,
                              hipStream_t stream) {
    (void)in_sizes; (void)n_in; (void)out_size; (void)ws_size;
    // setup_inputs order: hidden_state(0, unused), obs1(1, unused), obs2(2), W(3), b(4)
    const float* obs2 = (const float*)d_in[2];
    const float* W    = (const float*)d_in[3];
    const float* b    = (const float*)d_in[4];
    float* out  = (float*)d_out;
    float* grid = (float*)d_ws;                 // NPED*NN = 294912 floats (1.18 MB)

    const int gridElems = NPED * NN;
    zero_kernel<<<(gridElems + 255) / 256, 256, 0, stream>>>(grid, gridElems);

    dim3 g1(NPED / T1, J_BLOCKS);               // 32 x 8 blocks, 256 thr each
    occupancy_kernel<<<g1, T1, 0, stream>>>(obs2, grid);

    // 512 M-tiles * 8 N-tiles = 4096 waves; 8 waves/block -> 512 blocks
    const int waves = (NPED / 16) * (HID / 16);
    gemm_bias_wmma<<<(waves * 32) / 256, 256, 0, stream>>>(grid, W, b, out);
}